// GraphAttentionEncoderLayer_68264210202883
// MI455X (gfx1250) — compile-verified
//
#include <hip/hip_runtime.h>
#include <hip/hip_bf16.h>
#include <math.h>

#define N_NODES 50000
#define N_EDGES 800000
#define D_MODEL 128
#define N_HEADS 8
#define D_HEAD  16

typedef float v2f __attribute__((ext_vector_type(2)));
typedef float v8f __attribute__((ext_vector_type(8)));

// ---------------------------------------------------------------------------
// CDNA5 async memory->LDS copy (GLOBAL_LOAD_ASYNC_TO_LDS_B32, ASYNCcnt).
// GVS mode: mem = SADDR(64) + VADDR(32-bit byte offset); LDS dest address
// comes from a VGPR (low 32 bits of the generic shared pointer = LDS offset).
// ---------------------------------------------------------------------------
__device__ __forceinline__ void async_ld_lds_b32(unsigned lds_addr,
                                                 const float* __restrict__ base,
                                                 unsigned byte_off)
{
    asm volatile("global_load_async_to_lds_b32 %0, %1, %2"
                 :
                 : "v"(lds_addr), "v"(byte_off), "s"(base)
                 : "memory");
}
__device__ __forceinline__ void wait_async0()
{
    asm volatile("s_wait_asynccnt 0" ::: "memory");
}
__device__ __forceinline__ unsigned lds_off(const void* p)
{
    return (unsigned)(uintptr_t)p;   // LDS aperture: addr[31:0] is LDS offset
}

// ---------------------------------------------------------------------------
// Kernel 1: fused LayerNorm + QKV projection using V_WMMA_F32_16X16X4_F32.
// Block = 128 threads (4 wave32), handles a 16-row node tile.
// ---------------------------------------------------------------------------
__global__ __launch_bounds__(128)
void ln_qkv_kernel(const float* __restrict__ x,
                   const float* __restrict__ qkv_w,   // [128][384] row-major
                   const float* __restrict__ qkv_b,   // [384]
                   const float* __restrict__ ln_g,
                   const float* __restrict__ ln_b,
                   float* __restrict__ q,
                   float* __restrict__ k,
                   float* __restrict__ v)
{
    __shared__ float s_h[16][D_MODEL];
    __shared__ float s_mu[16], s_rs[16];

    const int tid  = threadIdx.x;
    const int row0 = blockIdx.x * 16;

    // Async-stage the 16x128 x-tile straight into LDS (no VGPR round-trip).
    #pragma unroll 4
    for (int r = 0; r < 16; ++r)
        async_ld_lds_b32(lds_off(&s_h[r][tid]), x,
                         (unsigned)(((size_t)(row0 + r) * D_MODEL + tid) * 4));
    wait_async0();
    __syncthreads();

    // LayerNorm stats: 8 lanes per row, wave32 shfl reduction.
    {
        const int r  = tid >> 3;
        const int l8 = tid & 7;
        float sum = 0.f, sq = 0.f;
        #pragma unroll
        for (int c = 0; c < 16; ++c) {
            float t = s_h[r][l8 * 16 + c];
            sum += t; sq += t * t;
        }
        #pragma unroll
        for (int off = 4; off; off >>= 1) {
            sum += __shfl_xor(sum, off, 32);
            sq  += __shfl_xor(sq,  off, 32);
        }
        if (l8 == 0) {
            float mu  = sum * (1.f / D_MODEL);
            float var = sq * (1.f / D_MODEL) - mu * mu;
            s_mu[r] = mu;
            s_rs[r] = rsqrtf(var + 1e-5f);
        }
    }
    __syncthreads();

    // Normalize in LDS.
    {
        const float g = ln_g[tid], bb = ln_b[tid];
        #pragma unroll 4
        for (int r = 0; r < 16; ++r)
            s_h[r][tid] = (s_h[r][tid] - s_mu[r]) * s_rs[r] * g + bb;
    }
    __syncthreads();

    // WMMA GEMM: 384 output cols = 24 tiles of 16; wave w owns tiles [6w, 6w+6).
    const int wave = tid >> 5;
    const int lane = tid & 31;
    const int n    = lane & 15;   // N (and A's M) index
    const int hi   = lane >> 4;   // which half-wave

    for (int t = wave * 6; t < wave * 6 + 6; ++t) {
        const int colbase = t * 16;
        v8f acc = {};
        #pragma unroll
        for (int kk = 0; kk < D_MODEL; kk += 4) {
            v2f a, b;
            a.x = s_h[n][kk + 2 * hi];
            a.y = s_h[n][kk + 2 * hi + 1];
            b.x = qkv_w[(size_t)(kk + 2 * hi)     * 384 + colbase + n];
            b.y = qkv_w[(size_t)(kk + 2 * hi + 1) * 384 + colbase + n];
            acc = __builtin_amdgcn_wmma_f32_16x16x4_f32(
                false, a, false, b, (short)0, acc, false, false);
        }
        const float bias = qkv_b[colbase + n];
        const int   buf  = colbase >> 7;            // 0=q 1=k 2=v
        const int   c0   = (colbase & 127) + n;
        float* dstp = (buf == 0) ? q : (buf == 1) ? k : v;
        #pragma unroll
        for (int i = 0; i < 8; ++i) {
            const int r = i + 8 * hi;               // C/D layout: VGPR i -> M = i + 8*hi
            dstp[(size_t)(row0 + r) * D_MODEL + c0] = acc[i] + bias;
        }
    }
}

// ---------------------------------------------------------------------------
// Kernel 2: zero/neg-inf init of segment buffers.
// ---------------------------------------------------------------------------
__global__ void init_kernel(float* __restrict__ m, float* __restrict__ s,
                            float* __restrict__ agg)
{
    const int i = blockIdx.x * blockDim.x + threadIdx.x;
    if (i < N_NODES * D_MODEL) agg[i] = 0.f;
    if (i < N_NODES * N_HEADS) { m[i] = -INFINITY; s[i] = 0.f; }
}

// ---------------------------------------------------------------------------
// Kernel 3: per-(edge,head) attention scores + segment max via float atomics.
// edge_attr read is the HBM-dominant stream; q/k gathers hit L2 (76.8MB<192MB).
// ---------------------------------------------------------------------------
__global__ __launch_bounds__(256)
void score_kernel(const float* __restrict__ q, const float* __restrict__ k,
                  const float* __restrict__ ea, const long long* __restrict__ ei,
                  float* __restrict__ scores, float* __restrict__ m)
{
    const int t = blockIdx.x * blockDim.x + threadIdx.x;
    if (t >= N_EDGES * N_HEADS) return;
    const int e = t >> 3, h = t & 7;
    const int src = (int)ei[e];
    const int dst = (int)ei[N_EDGES + e];

    const float4* qp = (const float4*)(q  + (size_t)dst * D_MODEL + h * D_HEAD);
    const float4* kp = (const float4*)(k  + (size_t)src * D_MODEL + h * D_HEAD);
    const float4* ep = (const float4*)(ea + (size_t)e   * D_MODEL + h * D_HEAD);

    float acc = 0.f;
    #pragma unroll
    for (int i = 0; i < 4; ++i) {
        const float4 qv = qp[i], kv = kp[i], ev = ep[i];
        acc += qv.x * (kv.x + ev.x) + qv.y * (kv.y + ev.y)
             + qv.z * (kv.z + ev.z) + qv.w * (kv.w + ev.w);
    }
    acc *= 0.25f;                                   // D_HEAD^-0.5 = 1/4
    scores[t] = acc;
    __hip_atomic_fetch_max(&m[dst * N_HEADS + h], acc,
                           __ATOMIC_RELAXED, __HIP_MEMORY_SCOPE_AGENT);
}

// ---------------------------------------------------------------------------
// Kernel 4: e = exp(score - m[dst]); segment-sum via atomicAdd.
// (m is finite for every node that has incoming edges, so the reference's
//  non-finite fixup can never be observed.)
// ---------------------------------------------------------------------------
__global__ __launch_bounds__(256)
void expsum_kernel(const long long* __restrict__ ei, float* __restrict__ scores,
                   const float* __restrict__ m, float* __restrict__ s)
{
    const int t = blockIdx.x * blockDim.x + threadIdx.x;
    if (t >= N_EDGES * N_HEADS) return;
    const int e = t >> 3, h = t & 7;
    const int dst = (int)ei[N_EDGES + e];
    const float val = expf(scores[t] - m[dst * N_HEADS + h]);
    scores[t] = val;
    atomicAdd(&s[dst * N_HEADS + h], val);
}

// ---------------------------------------------------------------------------
// Kernel 5: msg = v[src] * attn; scatter-add into agg[dst].
// ---------------------------------------------------------------------------
__global__ __launch_bounds__(256)
void aggregate_kernel(const long long* __restrict__ ei,
                      const float* __restrict__ scores, const float* __restrict__ s,
                      const float* __restrict__ v, float* __restrict__ agg)
{
    const int t = blockIdx.x * blockDim.x + threadIdx.x;
    if (t >= N_EDGES * N_HEADS) return;
    const int e = t >> 3, h = t & 7;
    const int src = (int)ei[e];
    const int dst = (int)ei[N_EDGES + e];

    const float attn = scores[t] / fmaxf(s[dst * N_HEADS + h], 1e-16f);
    const float4* vp = (const float4*)(v + (size_t)src * D_MODEL + h * D_HEAD);
    float* ap = agg + (size_t)dst * D_MODEL + h * D_HEAD;
    #pragma unroll
    for (int i = 0; i < 4; ++i) {
        const float4 vv = vp[i];
        atomicAdd(ap + 4 * i + 0, vv.x * attn);
        atomicAdd(ap + 4 * i + 1, vv.y * attn);
        atomicAdd(ap + 4 * i + 2, vv.z * attn);
        atomicAdd(ap + 4 * i + 3, vv.w * attn);
    }
}

// ---------------------------------------------------------------------------
// Kernel 6: out = agg @ out_w + out_b + x   (WMMA f32, residual fused).
// ---------------------------------------------------------------------------
__global__ __launch_bounds__(128)
void out_gemm_kernel(const float* __restrict__ agg,
                     const float* __restrict__ out_w,   // [128][128]
                     const float* __restrict__ out_b,
                     const float* __restrict__ x,
                     float* __restrict__ out)
{
    __shared__ float s_a[16][D_MODEL];
    const int tid  = threadIdx.x;
    const int row0 = blockIdx.x * 16;

    // Async-stage agg tile into LDS.
    #pragma unroll 4
    for (int r = 0; r < 16; ++r)
        async_ld_lds_b32(lds_off(&s_a[r][tid]), agg,
                         (unsigned)(((size_t)(row0 + r) * D_MODEL + tid) * 4));
    wait_async0();
    __syncthreads();

    const int wave = tid >> 5;
    const int lane = tid & 31;
    const int n    = lane & 15;
    const int hi   = lane >> 4;

    for (int t = wave * 2; t < wave * 2 + 2; ++t) {
        const int colbase = t * 16;
        v8f acc = {};
        #pragma unroll
        for (int kk = 0; kk < D_MODEL; kk += 4) {
            v2f a, b;
            a.x = s_a[n][kk + 2 * hi];
            a.y = s_a[n][kk + 2 * hi + 1];
            b.x = out_w[(size_t)(kk + 2 * hi)     * D_MODEL + colbase + n];
            b.y = out_w[(size_t)(kk + 2 * hi + 1) * D_MODEL + colbase + n];
            acc = __builtin_amdgcn_wmma_f32_16x16x4_f32(
                false, a, false, b, (short)0, acc, false, false);
        }
        const float bias = out_b[colbase + n];
        #pragma unroll
        for (int i = 0; i < 8; ++i) {
            const int r = i + 8 * hi;
            const size_t idx = (size_t)(row0 + r) * D_MODEL + colbase + n;
            out[idx] = x[idx] + acc[i] + bias;
        }
    }
}

// ---------------------------------------------------------------------------
extern "C" void kernel_launch(void* const* d_in, const int* in_sizes, int n_in,
                              void* d_out, int out_size, void* d_ws, size_t ws_size,
                              hipStream_t stream)
{
    const float*     x      = (const float*)d_in[0];
    const float*     ea     = (const float*)d_in[1];
    const float*     qkv_w  = (const float*)d_in[2];
    const float*     qkv_b  = (const float*)d_in[3];
    const float*     out_w  = (const float*)d_in[4];
    const float*     out_b  = (const float*)d_in[5];
    const float*     ln_g   = (const float*)d_in[6];
    const float*     ln_b   = (const float*)d_in[7];
    const long long* ei     = (const long long*)d_in[8];
    float*           out    = (float*)d_out;

    // Workspace layout (floats).
    float* ws = (float*)d_ws;
    const size_t NQ = (size_t)N_NODES * D_MODEL;          // 6.4M
    const size_t NS = (size_t)N_EDGES * N_HEADS;          // 6.4M
    const size_t NM = (size_t)N_NODES * N_HEADS;          // 0.4M
    float* q      = ws;                  ws += NQ;
    float* k      = ws;                  ws += NQ;
    float* v      = ws;                  ws += NQ;
    float* scores = ws;                  ws += NS;
    float* m      = ws;                  ws += NM;
    float* ssum   = ws;                  ws += NM;
    float* agg    = ws;                  /* ws += NQ; */

    const int rowTiles = N_NODES / 16;                    // 3125 exactly
    const int edgeBlk  = (N_EDGES * N_HEADS + 255) / 256; // 25000
    const int initBlk  = (N_NODES * D_MODEL + 255) / 256;

    ln_qkv_kernel   <<<rowTiles, 128, 0, stream>>>(x, qkv_w, qkv_b, ln_g, ln_b, q, k, v);
    init_kernel     <<<initBlk, 256, 0, stream>>>(m, ssum, agg);
    score_kernel    <<<edgeBlk, 256, 0, stream>>>(q, k, ea, ei, scores, m);
    expsum_kernel   <<<edgeBlk, 256, 0, stream>>>(ei, scores, m, ssum);
    aggregate_kernel<<<edgeBlk, 256, 0, stream>>>(ei, scores, ssum, v, agg);
    out_gemm_kernel <<<rowTiles, 128, 0, stream>>>(agg, out_w, out_b, x, out);
}